// NPNeighbors_61787399520647
// MI455X (gfx1250) — compile-verified
//
#include <hip/hip_runtime.h>
#include <hip/hip_bf16.h>
#include <math.h>

// Problem constants (from reference setup_inputs)
#define MM 8
#define NN 400
#define KK 27          // (2*N_IMAGES+1)^3
#define TILES 25       // 400 / 16
#define CUT2 36.0f     // CUTOFF^2

typedef float v2f __attribute__((ext_vector_type(2)));
typedef float v8f __attribute__((ext_vector_type(8)));

// ---------------------------------------------------------------------------
// Kernel 1: per-system prologue. Computes inv(cell), wrapped coords, packed
// wrap offsets (as float: 9*wx+3*wy+wz), and the 27 image shift vectors.
// ---------------------------------------------------------------------------
__global__ __launch_bounds__(64)
void np_prep_kernel(const float* __restrict__ coords,   // [M,N,3]
                    const float* __restrict__ cell,     // [M,3,3]
                    float* __restrict__ wrappedW,       // [M,N,3]
                    float* __restrict__ woF,            // [M,N] packed base-3 combo
                    float* __restrict__ shiftW)         // [M,K,3]
{
    const int m = blockIdx.x;
    const float* C = cell + m * 9;
    const float c00=C[0], c01=C[1], c02=C[2];
    const float c10=C[3], c11=C[4], c12=C[5];
    const float c20=C[6], c21=C[7], c22=C[8];
    const float det = c00*(c11*c22 - c12*c21)
                    - c01*(c10*c22 - c12*c20)
                    + c02*(c10*c21 - c11*c20);
    const float id = 1.0f / det;
    // inv[i][j] = cofactor(j,i)/det   (cell @ inv = I)
    const float i00 =  (c11*c22 - c12*c21)*id;
    const float i01 = -(c01*c22 - c02*c21)*id;
    const float i02 =  (c01*c12 - c02*c11)*id;
    const float i10 = -(c10*c22 - c12*c20)*id;
    const float i11 =  (c00*c22 - c02*c20)*id;
    const float i12 = -(c00*c12 - c02*c10)*id;
    const float i20 =  (c10*c21 - c11*c20)*id;
    const float i21 = -(c00*c21 - c01*c20)*id;
    const float i22 =  (c00*c11 - c01*c10)*id;

    for (int n = threadIdx.x; n < NN; n += blockDim.x) {
        const float* p = coords + (m*NN + n)*3;
        const float x = p[0], y = p[1], z = p[2];
        // proj_e = sum_d coord_d * inv[d][e]
        const float p0 = x*i00 + y*i10 + z*i20;
        const float p1 = x*i01 + y*i11 + z*i21;
        const float p2 = x*i02 + y*i12 + z*i22;
        const float f0 = floorf(p0), f1 = floorf(p1), f2 = floorf(p2);
        // wrapped_d = coord_d - sum_e f_e * cell[e][d]
        float* w = wrappedW + (m*NN + n)*3;
        w[0] = x - (f0*c00 + f1*c10 + f2*c20);
        w[1] = y - (f0*c01 + f1*c11 + f2*c21);
        w[2] = z - (f0*c02 + f1*c12 + f2*c22);
        // packed base-3 combo, kept in float (exact; |values| << 2^24)
        woF[m*NN + n] = 9.0f*f0 + 3.0f*f1 + f2;
    }
    if (threadIdx.x < KK) {
        const int k = threadIdx.x;
        const float kx = (float)(k/9 - 1);
        const float ky = (float)((k/3)%3 - 1);
        const float kz = (float)(k%3 - 1);
        float* s = shiftW + (m*KK + k)*3;
        // shift_d = sum_c img_c * cell[c][d]
        s[0] = kx*c00 + ky*c10 + kz*c20;
        s[1] = kx*c01 + ky*c11 + kz*c21;
        s[2] = kx*c02 + ky*c12 + kz*c22;
    }
}

// ---------------------------------------------------------------------------
// Kernel 2: one wave32 per 16x16 (i,j) tile of one (m,k) slab.
// d^2 via V_WMMA_F32_16X16X4_F32 Gram trick:
//   A row = (ax, ay, az, |a|^2),  a = wrapped_i + shift_k
//   B col = (-2bx, -2by, -2bz, 1), b = wrapped_j
//   C     = |b_j|^2 per column   =>  D = |a_i - b_j|^2
// offset_index = k + woPacked_j - woPacked_i  (k IS the base-3 digit code).
// ---------------------------------------------------------------------------
__global__ __launch_bounds__(256)
void np_pairs_kernel(const float* __restrict__ wrappedW,
                     const float* __restrict__ woF,
                     const float* __restrict__ shiftW,
                     float* __restrict__ out)
{
    const int wave = threadIdx.x >> 5;
    const int lane = threadIdx.x & 31;
    const int TOTAL = MM * KK * TILES * TILES;

    const int t0 = blockIdx.x * (blockDim.x >> 5) + wave;
    if (t0 >= TOTAL) return;                 // wave-uniform
    // wave-uniform: force the tile decomposition onto SALU
    int t = __builtin_amdgcn_readfirstlane(t0);

    const int tj = t % TILES;  t /= TILES;
    const int ti = t % TILES;  t /= TILES;
    const int k  = t % KK;
    const int m  = t / KK;
    const int i0 = ti * 16, j0 = tj * 16;

    const int  q  = lane & 15;
    const bool hi = lane >= 16;

    // --- A side: row (i0+q), shifted by image k ---
    const float* wa = wrappedW + (m*NN + i0 + q)*3;
    const float* sv = shiftW   + (m*KK + k)*3;
    const float ax = wa[0] + sv[0];
    const float ay = wa[1] + sv[1];
    const float az = wa[2] + sv[2];
    const float na2 = ax*ax + ay*ay + az*az;
    const float waC = woF[m*NN + i0 + q];    // 9*wax+3*way+waz (float, exact)

    // --- B side: column (j0+q) ---
    const float* wb = wrappedW + (m*NN + j0 + q)*3;
    const float bx = wb[0], by = wb[1], bz = wb[2];
    const float nb2 = bx*bx + by*by + bz*bz;
    const float wbC = woF[m*NN + j0 + q];

    // offset_index = k + wbC - waC_row  (k is already the base-3 digit code)
    const float offBase = (float)k + wbC;

    // --- WMMA operands per the 16x16x4 f32 VGPR layouts ---
    v2f A;  // A 16x4: V0 = K0 (lo) / K2 (hi), V1 = K1 (lo) / K3 (hi)
    A[0] = hi ? az  : ax;
    A[1] = hi ? na2 : ay;
    v2f B;  // B 4x16: V0 = row0 (lo) / row2 (hi), V1 = row1 (lo) / row3 (hi)
    B[0] = hi ? (-2.0f*bz) : (-2.0f*bx);
    B[1] = hi ? 1.0f       : (-2.0f*by);
    v8f Cc;
#pragma unroll
    for (int r = 0; r < 8; ++r) Cc[r] = nb2;   // |b_j|^2 broadcast per column

    v8f D = __builtin_amdgcn_wmma_f32_16x16x4_f32(
        false, A, false, B, (short)0, Cc, false, false);

    const size_t S = (size_t)MM * KK * NN * NN;   // 34,560,000
    const int j = j0 + q;
    const int rowBase = i0 + (hi ? 8 : 0);        // row of D VGPR0 for this lane
    const size_t idx0 = (((size_t)(m*KK + k)*NN + rowBase)*NN) + j;

    float* __restrict__ pd  = out + idx0;             // dist
    float* __restrict__ pp  = out + S + idx0*3;       // paircoord (x,y,z)
    float* __restrict__ pmk = out + 4*S + idx0;       // mask
    float* __restrict__ po  = out + 5*S + idx0;       // offset_index

    // A diagonal element can only occur when k==13 AND ti==tj; within such a
    // tile, i==j  <=>  srcl == q.  (wave-uniform scalar `anyDiag`)
    const bool anyDiag = (k == 13) & (ti == tj);

#pragma unroll
    for (int r = 0; r < 8; ++r) {
        const int srcl = r + (hi ? 8 : 0);      // lane holding row i0+srcl data
        const float axr = __shfl(ax, srcl, 32);
        const float ayr = __shfl(ay, srcl, 32);
        const float azr = __shfl(az, srcl, 32);
        const float wcr = __shfl(waC, srcl, 32);

        const float d2 = D[r];
        // flat predicate math (no short-circuit -> no exec-mask branching)
        const bool msk = (d2 < CUT2) & !(anyDiag & (srcl == q));
        const float sq = __builtin_amdgcn_sqrtf(fmaxf(d2, 0.0f));  // v_sqrt_f32

        __builtin_nontemporal_store(msk ? sq : 0.0f,          pd + (size_t)r*NN);
        __builtin_nontemporal_store(msk ? (axr - bx) : 0.0f,  pp + (size_t)r*NN*3 + 0);
        __builtin_nontemporal_store(msk ? (ayr - by) : 0.0f,  pp + (size_t)r*NN*3 + 1);
        __builtin_nontemporal_store(msk ? (azr - bz) : 0.0f,  pp + (size_t)r*NN*3 + 2);
        __builtin_nontemporal_store(msk ? 1.0f : 0.0f,        pmk + (size_t)r*NN);
        __builtin_nontemporal_store(offBase - wcr,            po  + (size_t)r*NN);
    }
}

// ---------------------------------------------------------------------------
extern "C" void kernel_launch(void* const* d_in, const int* in_sizes, int n_in,
                              void* d_out, int out_size, void* d_ws, size_t ws_size,
                              hipStream_t stream) {
    const float* coords = (const float*)d_in[0];   // [8,400,3] f32
    const float* cell   = (const float*)d_in[1];   // [8,3,3]   f32

    float* wrappedW = (float*)d_ws;                      // M*N*3 floats
    float* woFp     = wrappedW + MM*NN*3;                // M*N floats (packed wrap combo)
    float* shiftW   = woFp + MM*NN;                      // M*K*3 floats

    np_prep_kernel<<<MM, 64, 0, stream>>>(coords, cell, wrappedW, woFp, shiftW);

    const int TOTAL = MM * KK * TILES * TILES;           // 135,000 tiles
    const int wavesPerBlock = 8;                         // 256 threads
    const int blocks = (TOTAL + wavesPerBlock - 1) / wavesPerBlock;
    np_pairs_kernel<<<blocks, 256, 0, stream>>>(wrappedW, woFp, shiftW, (float*)d_out);
}